// GCN_55516747268431
// MI455X (gfx1250) — compile-verified
//
#include <hip/hip_runtime.h>
#include <hip/hip_bf16.h>

typedef float v2f __attribute__((ext_vector_type(2)));
typedef float v8f __attribute__((ext_vector_type(8)));

#define N_NODES 100000
#define N_EDGES 1600000
#define FEAT_IN 128
#define FEAT_HID 128
#define FEAT_OUT 64

// ---------------------------------------------------------------------------
// Normalization precompute
// ---------------------------------------------------------------------------
__global__ void k_fill_deg(float* __restrict__ deg, int n) {
    int i = blockIdx.x * blockDim.x + threadIdx.x;
    if (i < n) deg[i] = 1.0f;  // self-loop weight 1 folded into init
}

__global__ void k_deg_scatter(const long long* __restrict__ ei,
                              const float* __restrict__ ew,
                              float* __restrict__ deg, int e) {
    int i = blockIdx.x * blockDim.x + threadIdx.x;
    if (i < e) {
        int d = (int)ei[(size_t)e + i];  // dst row of edge_index
        unsafeAtomicAdd(&deg[d], ew[i]);
    }
}

__global__ void k_dinv(const float* __restrict__ deg,
                       float* __restrict__ dinv,
                       float* __restrict__ selfn, int n) {
    int i = blockIdx.x * blockDim.x + threadIdx.x;
    if (i < n) {
        float d = deg[i];
        float r = (d > 0.0f) ? rsqrtf(d) : 0.0f;
        dinv[i] = r;
        selfn[i] = r * r;  // norm of the self-loop edge
    }
}

__global__ void k_norm(const long long* __restrict__ ei,
                       const float* __restrict__ ew,
                       const float* __restrict__ dinv,
                       float* __restrict__ norm, int e) {
    int i = blockIdx.x * blockDim.x + threadIdx.x;
    if (i < e) {
        int s = (int)ei[i];
        int d = (int)ei[(size_t)e + i];
        norm[i] = dinv[s] * ew[i] * dinv[d];
    }
}

// ---------------------------------------------------------------------------
// f32 WMMA GEMM:  C[N x FOUT] = A[N x 128] * W[128 x FOUT]
// One wave per 16x16 output tile; K unrolled in steps of 4.
// ---------------------------------------------------------------------------
template <int FOUT>
__global__ void k_gemm_wmma(const float* __restrict__ A,
                            const float* __restrict__ W,
                            float* __restrict__ C) {
    const int K = 128;
    const int colTiles = FOUT / 16;
    int lane = threadIdx.x & 31;
    int wave = threadIdx.x >> 5;
    int tile = blockIdx.x * (blockDim.x >> 5) + wave;
    int row0 = (tile / colTiles) * 16;
    int col0 = (tile % colTiles) * 16;

    int m  = lane & 15;   // M (for A) / N (for B) index of this lane
    int kh = lane >> 4;   // half-wave selects K pair

    const float* arow = A + (size_t)(row0 + m) * K;   // row-major A row
    const float* bcol = W + col0 + m;                 // column of W

    v8f acc = {};
#pragma unroll
    for (int k0 = 0; k0 < K; k0 += 4) {
        // A 16x4 fragment: VGPR0/1 hold K = 2*kh, 2*kh+1 (8B-aligned pair load)
        v2f a = *(const v2f*)(arow + k0 + 2 * kh);
        // B 4x16 fragment: VGPR0/1 hold K = 2*kh, 2*kh+1 for column col0+m
        v2f b;
        b.x = bcol[(size_t)(k0 + 2 * kh) * FOUT];
        b.y = bcol[(size_t)(k0 + 2 * kh + 1) * FOUT];
        acc = __builtin_amdgcn_wmma_f32_16x16x4_f32(
            /*neg_a=*/false, a, /*neg_b=*/false, b,
            /*c_mod=*/(short)0, acc, /*reuse_a=*/false, /*reuse_b=*/false);
    }

    // C/D layout: VGPR j -> row j (lanes 0-15) / row j+8 (lanes 16-31), col = lane&15
    int rbase = row0 + 8 * kh;
    int col = col0 + m;
#pragma unroll
    for (int j = 0; j < 8; ++j) {
        C[(size_t)(rbase + j) * FOUT + col] = acc[j];
    }
}

// ---------------------------------------------------------------------------
// out[i][:] = selfnorm[i] * h[i][:]   (initializes aggregation buffer)
// ---------------------------------------------------------------------------
template <int F>
__global__ void k_selfinit(const float* __restrict__ h,
                           const float* __restrict__ selfn,
                           float* __restrict__ out, int total4) {
    int i = blockIdx.x * blockDim.x + threadIdx.x;
    if (i < total4) {
        int row = i / (F / 4);
        float s = selfn[row];
        float4 v = ((const float4*)h)[i];
        float4 o;
        o.x = s * v.x; o.y = s * v.y; o.z = s * v.z; o.w = s * v.w;
        ((float4*)out)[i] = o;
    }
}

// ---------------------------------------------------------------------------
// Edge scatter: out[dst] += norm[e] * h[src]  (wave per edge, lane-vectorized)
// All feature rows are L2-resident (51 MB << 192 MB), so f32 atomics resolve
// in L2 at near-L2 bandwidth.
// ---------------------------------------------------------------------------
template <int F>
__global__ void k_edge_scatter(const float* __restrict__ h,
                               const long long* __restrict__ ei,
                               const float* __restrict__ norm,
                               float* __restrict__ out, int e) {
    const int VPL = F / 32;  // floats per lane (4 for F=128, 2 for F=64)
    int lane = threadIdx.x & 31;
    int wave = threadIdx.x >> 5;
    long long eidx = (long long)blockIdx.x * (blockDim.x >> 5) + wave;
    if (eidx >= e) return;
    int s = (int)ei[eidx];
    int d = (int)ei[(size_t)e + eidx];
    float nv = norm[eidx];
    const float* hs = h + (size_t)s * F + lane * VPL;
    float* od = out + (size_t)d * F + lane * VPL;
#pragma unroll
    for (int j = 0; j < VPL; ++j) {
        unsafeAtomicAdd(&od[j], nv * hs[j]);
    }
}

// ---------------------------------------------------------------------------
// x[i] = relu(x[i] + b[i % F])   (F is a power of two)
// ---------------------------------------------------------------------------
template <int F, bool RELU>
__global__ void k_bias_act(float* __restrict__ x, const float* __restrict__ b,
                           int total) {
    int i = blockIdx.x * blockDim.x + threadIdx.x;
    if (i < total) {
        float v = x[i] + b[i & (F - 1)];
        if (RELU) v = fmaxf(v, 0.0f);
        x[i] = v;
    }
}

// ---------------------------------------------------------------------------
// In-place: out[row][:] = log_softmax(out[row][:] + b3), F_OUT = 64.
// Wave per row; two elements per lane; wave32 shuffle reductions.
// ---------------------------------------------------------------------------
__global__ void k_logsoftmax64(float* __restrict__ out,
                               const float* __restrict__ bias, int n) {
    int lane = threadIdx.x & 31;
    int wave = threadIdx.x >> 5;
    int row = blockIdx.x * (blockDim.x >> 5) + wave;
    if (row >= n) return;
    float* p = out + (size_t)row * 64;
    float v0 = p[lane] + bias[lane];
    float v1 = p[lane + 32] + bias[lane + 32];
    float m = fmaxf(v0, v1);
#pragma unroll
    for (int o = 16; o > 0; o >>= 1) m = fmaxf(m, __shfl_xor(m, o, 32));
    float s = expf(v0 - m) + expf(v1 - m);
#pragma unroll
    for (int o = 16; o > 0; o >>= 1) s += __shfl_xor(s, o, 32);
    float lg = m + logf(s);
    p[lane] = v0 - lg;
    p[lane + 32] = v1 - lg;
}

// ---------------------------------------------------------------------------
extern "C" void kernel_launch(void* const* d_in, const int* in_sizes, int n_in,
                              void* d_out, int out_size, void* d_ws,
                              size_t ws_size, hipStream_t stream) {
    const float* x      = (const float*)d_in[0];
    const long long* ei = (const long long*)d_in[1];  // [2, E] int64
    const float* ew     = (const float*)d_in[2];
    const float* W1     = (const float*)d_in[3];
    const float* b1     = (const float*)d_in[4];
    const float* W2     = (const float*)d_in[5];
    const float* b2     = (const float*)d_in[6];
    const float* W3     = (const float*)d_in[7];
    const float* b3     = (const float*)d_in[8];
    float* out = (float*)d_out;

    // Workspace layout (floats): bufA | bufB | norm | deg | dinv | selfn
    float* ws = (float*)d_ws;
    size_t nf = (size_t)N_NODES * FEAT_HID;
    float* bufA  = ws;
    float* bufB  = bufA + nf;
    float* norm  = bufB + nf;
    float* deg   = norm + N_EDGES;
    float* dinv  = deg + N_NODES;
    float* selfn = dinv + N_NODES;

    const int T = 256;  // 8 waves per block (wave32)
    const int nBlkN = (N_NODES + T - 1) / T;
    const int nBlkE = (N_EDGES + T - 1) / T;

    // ---- gcn_norm ----
    k_fill_deg<<<nBlkN, T, 0, stream>>>(deg, N_NODES);
    k_deg_scatter<<<nBlkE, T, 0, stream>>>(ei, ew, deg, N_EDGES);
    k_dinv<<<nBlkN, T, 0, stream>>>(deg, dinv, selfn, N_NODES);
    k_norm<<<nBlkE, T, 0, stream>>>(ei, ew, dinv, norm, N_EDGES);

    const int edgeBlocks = (N_EDGES + 7) / 8;  // wave per edge

    // ---- layer 1: relu(agg(x @ W1) + b1) ----
    {
        int tiles = (N_NODES / 16) * (FEAT_HID / 16);  // 50000, /8 exact
        k_gemm_wmma<FEAT_HID><<<tiles / 8, T, 0, stream>>>(x, W1, bufA);
        int tot4 = (int)(nf / 4);
        k_selfinit<FEAT_HID><<<(tot4 + T - 1) / T, T, 0, stream>>>(bufA, selfn, bufB, tot4);
        k_edge_scatter<FEAT_HID><<<edgeBlocks, T, 0, stream>>>(bufA, ei, norm, bufB, N_EDGES);
        k_bias_act<FEAT_HID, true><<<((int)nf + T - 1) / T, T, 0, stream>>>(bufB, b1, (int)nf);
    }
    // ---- layer 2 ----
    {
        int tiles = (N_NODES / 16) * (FEAT_HID / 16);
        k_gemm_wmma<FEAT_HID><<<tiles / 8, T, 0, stream>>>(bufB, W2, bufA);
        int tot4 = (int)(nf / 4);
        // bufB (layer-1 activations) fully consumed by the GEMM above; reuse it
        k_selfinit<FEAT_HID><<<(tot4 + T - 1) / T, T, 0, stream>>>(bufA, selfn, bufB, tot4);
        k_edge_scatter<FEAT_HID><<<edgeBlocks, T, 0, stream>>>(bufA, ei, norm, bufB, N_EDGES);
        k_bias_act<FEAT_HID, true><<<((int)nf + T - 1) / T, T, 0, stream>>>(bufB, b2, (int)nf);
    }
    // ---- layer 3 + log_softmax (aggregate straight into d_out) ----
    {
        int tiles = (N_NODES / 16) * (FEAT_OUT / 16);  // 25000, /8 exact
        k_gemm_wmma<FEAT_OUT><<<tiles / 8, T, 0, stream>>>(bufB, W3, bufA);
        size_t no = (size_t)N_NODES * FEAT_OUT;
        int tot4 = (int)(no / 4);
        k_selfinit<FEAT_OUT><<<(tot4 + T - 1) / T, T, 0, stream>>>(bufA, selfn, out, tot4);
        k_edge_scatter<FEAT_OUT><<<edgeBlocks, T, 0, stream>>>(bufA, ei, norm, out, N_EDGES);
        k_logsoftmax64<<<(N_NODES + 7) / 8, T, 0, stream>>>(out, b3, N_NODES);
    }
}